// FeaturesLoss_56564719288689
// MI455X (gfx1250) — compile-verified
//
#include <hip/hip_runtime.h>
#include <hip/hip_bf16.h>

// CDNA5 / gfx1250 contrastive loss.
// Gram matrix via V_WMMA_F32_16X16X4_F32 (exact f32 path), tiled 128x128 per
// 256-thread (8-wave, wave32) block, full K=128 panel resident in LDS.

typedef __attribute__((ext_vector_type(2))) float v2f;
typedef __attribute__((ext_vector_type(8))) float v8f;

#define TILE        128
#define LDS_STRIDE  132   // padded: bank-conflict-free fragment loads, 16B aligned rows
#define D_DIM       128
#define B_TOTAL     8192
#define MARGIN      2.0f

__global__ void __launch_bounds__(256)
contrastive_init_kernel(float* accum, unsigned int* cnt) {
    if (threadIdx.x == 0) { *accum = 0.0f; *cnt = 0u; }
}

__global__ void __launch_bounds__(256)
contrastive_tile_kernel(const float* __restrict__ feat,
                        const int* __restrict__ labels,
                        float* __restrict__ loss_accum,
                        unsigned int* __restrict__ pos_accum) {
    const int ti = blockIdx.y;     // row tile
    const int tj = blockIdx.x;     // col tile
    if (tj < ti) return;           // symmetric: upper triangle only

    __shared__ float As[TILE * LDS_STRIDE];
    __shared__ float Bs[TILE * LDS_STRIDE];
    __shared__ float sqA[TILE], sqB[TILE];
    __shared__ int   labA[TILE], labB[TILE];
    __shared__ float wave_part[8];
    __shared__ unsigned int wave_cnt[8];

    const int tid    = threadIdx.x;
    const int lane   = tid & 31;
    const int wave   = tid >> 5;
    const int lane16 = lane & 15;
    const int khalf  = (lane >> 4) << 1;   // 0 for lanes 0-15, 2 for lanes 16-31

    const int rowbase = ti * TILE;
    const int colbase = tj * TILE;

    // ---- stage both 128x128 f32 feature panels into LDS (coalesced float4) ----
    for (int f = tid; f < TILE * (D_DIM / 4); f += 256) {
        const int r  = f >> 5;          // row 0..127
        const int c4 = f & 31;          // float4 column 0..31
        float4 a = ((const float4*)(feat + (size_t)(rowbase + r) * D_DIM))[c4];
        float4 b = ((const float4*)(feat + (size_t)(colbase + r) * D_DIM))[c4];
        *(float4*)(&As[r * LDS_STRIDE + c4 * 4]) = a;
        *(float4*)(&Bs[r * LDS_STRIDE + c4 * 4]) = b;
    }
    if (tid < TILE) {
        labA[tid] = labels[rowbase + tid];
        labB[tid] = labels[colbase + tid];
    }
    __syncthreads();

    // ---- per-row squared norms straight from LDS ----
    if (tid < TILE) {
        const float* rowp = &As[tid * LDS_STRIDE];
        float s = 0.0f;
        #pragma unroll
        for (int k = 0; k < D_DIM; k += 4) {
            float4 v = *(const float4*)(rowp + k);
            s += v.x * v.x + v.y * v.y + v.z * v.z + v.w * v.w;
        }
        sqA[tid] = s;
    } else if (tid < 2 * TILE) {
        const int r = tid - TILE;
        const float* rowp = &Bs[r * LDS_STRIDE];
        float s = 0.0f;
        #pragma unroll
        for (int k = 0; k < D_DIM; k += 4) {
            float4 v = *(const float4*)(rowp + k);
            s += v.x * v.x + v.y * v.y + v.z * v.z + v.w * v.w;
        }
        sqB[r] = s;
    }
    __syncthreads();

    // ---- wave tiling: 8 waves in 4(M) x 2(N); each wave owns 32x64 ----
    const int wm = wave & 3;   // 32-row strip
    const int wn = wave >> 2;  // 64-col strip

    v8f acc[2][4];
    #pragma unroll
    for (int sm = 0; sm < 2; ++sm)
        #pragma unroll
        for (int sn = 0; sn < 4; ++sn)
            acc[sm][sn] = (v8f){0.f, 0.f, 0.f, 0.f, 0.f, 0.f, 0.f, 0.f};

    // K loop: 32 steps of K=4, fragments are contiguous b64 LDS loads.
    for (int k0 = 0; k0 < D_DIM; k0 += 4) {
        const int kk = k0 + khalf;
        v2f a[2], b[4];
        #pragma unroll
        for (int sm = 0; sm < 2; ++sm) {
            const int row = wm * 32 + sm * 16 + lane16;
            a[sm] = *(const v2f*)(&As[row * LDS_STRIDE + kk]);
        }
        #pragma unroll
        for (int sn = 0; sn < 4; ++sn) {
            const int col = wn * 64 + sn * 16 + lane16;
            b[sn] = *(const v2f*)(&Bs[col * LDS_STRIDE + kk]);
        }
        #pragma unroll
        for (int sm = 0; sm < 2; ++sm)
            #pragma unroll
            for (int sn = 0; sn < 4; ++sn)
                acc[sm][sn] = __builtin_amdgcn_wmma_f32_16x16x4_f32(
                    false, a[sm], false, b[sn],
                    (short)0, acc[sm][sn], false, false);
    }

    // ---- epilogue: map C/D VGPR layout -> (i, j), accumulate loss ----
    const float wgt = (ti == tj) ? 1.0f : 2.0f;
    const unsigned int wgt_i = (ti == tj) ? 1u : 2u;
    const int rhi = (lane >> 4) << 3;   // +8 rows for lanes 16-31

    float part = 0.0f;
    unsigned int pcnt = 0u;
    #pragma unroll
    for (int sm = 0; sm < 2; ++sm) {
        #pragma unroll
        for (int sn = 0; sn < 4; ++sn) {
            #pragma unroll
            for (int r = 0; r < 8; ++r) {
                const int rl = wm * 32 + sm * 16 + r + rhi;
                const int cl = wn * 64 + sn * 16 + lane16;
                const int gi = rowbase + rl;
                const int gj = colbase + cl;
                if (gi == gj) continue;            // skip diagonal
                const float g  = acc[sm][sn][r];
                float d2 = sqA[rl] + sqB[cl] - 2.0f * g;
                d2 = fmaxf(d2, 0.0f);
                if (labA[rl] == labB[cl]) {
                    part += wgt * d2;
                    pcnt += wgt_i;
                } else {
                    const float d = sqrtf(d2);
                    const float h = fmaxf(MARGIN - d, 0.0f);
                    part += wgt * (h * h);
                }
            }
        }
    }

    // ---- wave32 shuffle reduction, then one atomic per block ----
    #pragma unroll
    for (int off = 16; off > 0; off >>= 1) {
        part += __shfl_down(part, off, 32);
        pcnt += (unsigned int)__shfl_down((int)pcnt, off, 32);
    }
    if (lane == 0) { wave_part[wave] = part; wave_cnt[wave] = pcnt; }
    __syncthreads();
    if (tid == 0) {
        float s = 0.0f;
        unsigned int c = 0u;
        #pragma unroll
        for (int i = 0; i < 8; ++i) { s += wave_part[i]; c += wave_cnt[i]; }
        atomicAdd(loss_accum, s);
        atomicAdd(pos_accum, c);
    }
}

__global__ void __launch_bounds__(256)
contrastive_final_kernel(const float* accum, const unsigned int* cnt, float* out) {
    if (threadIdx.x == 0) {
        const float loss = *accum;
        const float total_pairs = (float)B_TOTAL * (float)(B_TOTAL - 1); // exact in f32
        out[0] = (*cnt > 0u) ? (loss / total_pairs) : loss;
    }
}

extern "C" void kernel_launch(void* const* d_in, const int* in_sizes, int n_in,
                              void* d_out, int out_size, void* d_ws, size_t ws_size,
                              hipStream_t stream) {
    (void)in_sizes; (void)n_in; (void)out_size; (void)ws_size;
    const float* feat   = (const float*)d_in[0];   // [8192, 128] f32
    const int*   labels = (const int*)d_in[1];     // [8192] int32 (JAX x64 off)
    float*       out    = (float*)d_out;           // scalar

    float*        accum = (float*)d_ws;                              // 4 bytes
    unsigned int* cnt   = (unsigned int*)((char*)d_ws + sizeof(float)); // 4 bytes

    contrastive_init_kernel<<<dim3(1), dim3(256), 0, stream>>>(accum, cnt);

    dim3 grid(B_TOTAL / TILE, B_TOTAL / TILE);   // 64 x 64 tiles, upper triangle works
    contrastive_tile_kernel<<<grid, dim3(256), 0, stream>>>(feat, labels, accum, cnt);

    contrastive_final_kernel<<<dim3(1), dim3(256), 0, stream>>>(accum, cnt, out);
}